// BertCRFModel2_51144470561282
// MI455X (gfx1250) — compile-verified
//
#include <hip/hip_runtime.h>
#include <math.h>

typedef __attribute__((ext_vector_type(2))) float v2f;
typedef __attribute__((ext_vector_type(8))) float v8f;

#define Bdim 64
#define Tdim 512
#define Hdim 768
#define Kdim 13

// -----------------------------------------------------------------------------
// Kernel 1: logits = hidden_states @ W + b   via V_WMMA_F32_16X16X4_F32
// Each wave32 owns a 16-row tile of the (32768 x 13) output (cols padded to 16).
// A-frag (16x4 f32): lane = m + 16*(k>=2), vgpr = k&1  -> one b64 load / step.
// B-frag (4x16 f32): staged in LDS pair-interleaved so each step is one b64.
// -----------------------------------------------------------------------------
__global__ __launch_bounds__(256) void gemm_wmma_kernel(
    const float* __restrict__ hs,   // [B*T, H]
    const float* __restrict__ W,    // [H, K]
    const float* __restrict__ bias, // [K]
    float* __restrict__ out)        // d_out: [0]=loss, [1..]=logits (B*T*K)
{
    // W padded to 16 cols, pair-interleaved: idx = (h>>1)*32 + n*2 + (h&1)
    __shared__ float wlds[(Hdim / 2) * 32]; // 12288 floats = 48 KB
    const int tid = threadIdx.x;
    for (int idx = tid; idx < (Hdim / 2) * 32; idx += 256) {
        int g   = idx >> 5;
        int rem = idx & 31;
        int n   = rem >> 1;
        int p   = rem & 1;
        int h   = 2 * g + p;
        wlds[idx] = (n < Kdim) ? W[h * Kdim + n] : 0.0f;
    }
    __syncthreads();

    const int lane  = tid & 31;
    const int wave  = tid >> 5;
    const int m     = lane & 15;
    const int khalf = lane >> 4;
    const long rowBase = ((long)blockIdx.x * 8 + wave) * 16;

    const float* aptr = hs + (size_t)(rowBase + m) * Hdim + 2 * khalf;
    const float* bptr = wlds + khalf * 32 + (lane & 15) * 2;

    v8f c = {};
#pragma unroll 4
    for (int h0 = 0; h0 < Hdim; h0 += 4) {
        v2f a = *(const v2f*)(aptr + h0);                 // global_load_b64
        v2f bfrag = *(const v2f*)(bptr + (h0 >> 1) * 32); // ds_load_b64
        // (neg_a, A, neg_b, B, c_mod, C, reuse_a, reuse_b)
        c = __builtin_amdgcn_wmma_f32_16x16x4_f32(false, a, false, bfrag,
                                                  (short)0, c, false, false);
    }

    if (blockIdx.x == 0 && tid == 0) out[0] = 0.0f; // init loss accumulator

    const int n = lane & 15;
    if (n < Kdim) {
        float bn = bias[n];
#pragma unroll
        for (int r = 0; r < 8; ++r) {
            long row = rowBase + r + 8 * khalf; // C/D layout: M = r + 8*(lane/16)
            out[1 + row * Kdim + n] = c[r] + bn;
        }
    }
}

// -----------------------------------------------------------------------------
// Kernel 2: CRF negative log-likelihood. One wave32 per batch; lane j holds
// alpha[j] (j < 13). Inner logsumexp over i uses __shfl broadcasts; transitions
// column trans[:, j] lives in registers. Loss accumulated with atomicAdd.
// -----------------------------------------------------------------------------
__global__ __launch_bounds__(32) void crf_kernel(
    const int*   __restrict__ labels, // [B, T]
    const int*   __restrict__ mask,   // [B, T]
    const float* __restrict__ start,  // [K]
    const float* __restrict__ endt,   // [K]
    const float* __restrict__ trans,  // [K, K]
    float* __restrict__ out)          // [0]=loss, [1..]=logits
{
    const int b    = blockIdx.x;
    const int j    = threadIdx.x;
    const int jeff = (j < Kdim) ? j : (Kdim - 1);

    const float* em = out + 1 + (size_t)b * Tdim * Kdim;

    float tcol[Kdim]; // trans[i][j]
#pragma unroll
    for (int i = 0; i < Kdim; ++i) tcol[i] = trans[i * Kdim + jeff];

    float em0   = em[jeff];
    float alpha = start[jeff] + em0;

    int   prev = labels[b * Tdim];
    float num  = start[prev] + __shfl(em0, prev, 32);
    int   cnt  = mask[b * Tdim];

    for (int t = 1; t < Tdim; ++t) {
        float emt = em[t * Kdim + jeff];
        int   mt  = mask[b * Tdim + t];
        int   cur = labels[b * Tdim + t];

        // numerator (same scalar on all lanes)
        float tpc  = trans[prev * Kdim + cur];
        float emc  = __shfl(emt, cur, 32);
        num += (tpc + emc) * (float)mt;
        cnt += mt;
        prev = cur;

        // denominator forward step: nxt[j] = logsumexp_i(alpha[i]+trans[i][j]) + em[t][j]
        float s[Kdim];
        float mx = -1e30f;
#pragma unroll
        for (int i = 0; i < Kdim; ++i) {
            s[i] = __shfl(alpha, i, 32) + tcol[i];
            mx = fmaxf(mx, s[i]);
        }
        float sum = 0.0f;
#pragma unroll
        for (int i = 0; i < Kdim; ++i) sum += __expf(s[i] - mx);
        float nxt = mx + __logf(sum) + emt;
        alpha = mt ? nxt : alpha;
    }

    // numerator end-transition
    int seq_end = cnt - 1;
    num += endt[labels[b * Tdim + seq_end]];

    // denominator = logsumexp_j(alpha[j] + end[j])
    float v = (j < Kdim) ? (alpha + endt[jeff]) : -1e30f;
    float vs[Kdim];
    float mx = -1e30f;
#pragma unroll
    for (int i = 0; i < Kdim; ++i) {
        vs[i] = __shfl(v, i, 32);
        mx = fmaxf(mx, vs[i]);
    }
    float sum = 0.0f;
#pragma unroll
    for (int i = 0; i < Kdim; ++i) sum += __expf(vs[i] - mx);
    float den = mx + __logf(sum);

    if (j == 0) atomicAdd(out, den - num); // loss = sum(den - num)
}

// -----------------------------------------------------------------------------
extern "C" void kernel_launch(void* const* d_in, const int* in_sizes, int n_in,
                              void* d_out, int out_size, void* d_ws, size_t ws_size,
                              hipStream_t stream) {
    const float* hs     = (const float*)d_in[0]; // hidden_states (B,T,H) f32
    const int*   labels = (const int*)  d_in[1]; // labels (B,T)
    const int*   mask   = (const int*)  d_in[2]; // attention_mask (B,T)
    const float* W      = (const float*)d_in[3]; // (H,K)
    const float* bias   = (const float*)d_in[4]; // (K,)
    const float* start  = (const float*)d_in[5]; // (K,)
    const float* endt   = (const float*)d_in[6]; // (K,)
    const float* trans  = (const float*)d_in[7]; // (K,K)
    float* out = (float*)d_out;

    // 32768 rows / (8 waves * 16 rows) = 256 blocks of 256 threads
    dim3 grid1((Bdim * Tdim) / (16 * 8));
    gemm_wmma_kernel<<<grid1, 256, 0, stream>>>(hs, W, bias, out);

    // one wave32 per batch sequence
    crf_kernel<<<Bdim, 32, 0, stream>>>(labels, mask, start, endt, trans, out);
}